// MambaBlock_78786880078490
// MI455X (gfx1250) — compile-verified
//
#include <hip/hip_runtime.h>
#include <hip/hip_bf16.h>
#include <math.h>

// ---------------------------------------------------------------------------
// Mamba block stack for MI455X (gfx1250, wave32).
// L=2304 tokens, d_model=128, d_inner=256, d_state=16, depth=8.
// ~4.5 GFLOP, ~15MB working set -> L2-resident, latency bound; fp32 WMMA
// (V_WMMA_F32_16X16X4_F32) keeps reference precision with matrix-pipe GEMMs.
// This revision removes all EXEC-divergent guards from the WMMA K-loops
// (W_xproj is zero-padded to N=48 in workspace) and has each wave compute
// MT=4 M-tiles per N-tile so B operands are loaded once per K-step and feed
// 4 back-to-back WMMAs.
// ---------------------------------------------------------------------------

#define L_SEQ   2304
#define MTILES  (L_SEQ / 16)     // 144
#define DM      128
#define DI      256
#define DS      16
#define DTR     8
#define DCONV   4
#define DEPTH   8
#define DBL_N   40               // dt_rank + 2*d_state
#define DBL_LD  48               // padded row stride for dbl / padded W_xproj
#define EPS_RMS 1e-5f

typedef __attribute__((ext_vector_type(2))) float v2f;
typedef __attribute__((ext_vector_type(8))) float v8f;

// ---------------------------------------------------------------------------
// WMMA f32 16x16x4 operand layout (CDNA5 ISA 7.12.2):
//   A 16x4 : lanes 0-15 -> M=lane, VGPR v -> K=v ; lanes 16-31 -> K=v+2
//   B 4x16 : lo half rows K=0,1 / hi half K=2,3 per VGPR ; N = lane%16
//   C/D    : VGPR r -> M=r (lanes 0-15) / M=r+8 (lanes 16-31); N = lane%16
// EXEC is all ones throughout (no guards anywhere in the GEMM kernels).
// ---------------------------------------------------------------------------

// MT consecutive 16-row tiles at one 16-col tile; B loaded once per K-step.
template<int MT>
__device__ __forceinline__ void wmma_panel(const float* __restrict__ A, int lda,
                                           const float* __restrict__ B, int ldb,
                                           int K, int m0, int n0, v8f (&acc)[MT])
{
    const int lane = threadIdx.x & 31;
    const int lid  = lane & 15;
    const int koff = (lane >> 4) * 2;          // hi half covers K+2, K+3
    const float* __restrict__ Arow = A + (m0 + lid) * lda + koff;
    const float* __restrict__ Bcol = B + koff * ldb + (n0 + lid);
    for (int k = 0; k < K; k += 4) {
        v2f b;
        b.x = Bcol[(size_t)k * ldb];
        b.y = Bcol[(size_t)(k + 1) * ldb];
        #pragma unroll
        for (int t = 0; t < MT; ++t) {
            v2f a;
            a.x = Arow[(size_t)t * 16 * lda + k + 0];
            a.y = Arow[(size_t)t * 16 * lda + k + 1];
            acc[t] = __builtin_amdgcn_wmma_f32_16x16x4_f32(false, a, false, b,
                                                           (short)0, acc[t],
                                                           false, false);
        }
    }
}

__device__ __forceinline__ void wmma_store(float* __restrict__ D, int ldd,
                                           int m0, int n0, v8f acc)
{
    const int lane = threadIdx.x & 31;
    const int half = lane >> 4;
    const int col  = n0 + (lane & 15);
    #pragma unroll
    for (int r = 0; r < 8; ++r)
        D[(size_t)(m0 + r + half * 8) * ldd + col] = acc[r];
}

// Generic unguarded WMMA GEMM: one wave per (MT x 16)-row, 16-col panel.
template<int MT>
__global__ void k_gemm(const float* __restrict__ A, const float* __restrict__ B,
                       float* __restrict__ D,
                       int mgroups, int ntiles, int K, int lda, int ldb, int ldd)
{
    const int wave = (blockIdx.x * blockDim.x + threadIdx.x) >> 5;
    if (wave >= mgroups * ntiles) return;       // wave-uniform exit
    const int mg = wave / ntiles;
    const int nt = wave % ntiles;
    const int m0 = mg * MT * 16;
    v8f acc[MT] = {};
    wmma_panel<MT>(A, lda, B, ldb, K, m0, nt * 16, acc);
    #pragma unroll
    for (int t = 0; t < MT; ++t)
        wmma_store(D, ldd, m0 + t * 16, nt * 16, acc[t]);
}

// dt GEMM: dt = softplus(dt_r @ W_dt + b_dt). A = dbl cols 0..7 (lda=48), K=8.
template<int MT>
__global__ void k_gemm_dt(const float* __restrict__ dbl, const float* __restrict__ Wdt,
                          const float* __restrict__ bias, float* __restrict__ dt)
{
    const int ntiles = DI >> 4;                 // 16
    const int wave = (blockIdx.x * blockDim.x + threadIdx.x) >> 5;
    if (wave >= (MTILES / MT) * ntiles) return;
    const int mg = wave / ntiles;
    const int nt = wave % ntiles;
    const int m0 = mg * MT * 16;
    v8f acc[MT] = {};
    wmma_panel<MT>(dbl, DBL_LD, Wdt, DI, DTR, m0, nt * 16, acc);
    const int lane = threadIdx.x & 31;
    const int half = lane >> 4;
    const int col  = nt * 16 + (lane & 15);
    const float bb = bias[col];
    #pragma unroll
    for (int t = 0; t < MT; ++t)
        #pragma unroll
        for (int r = 0; r < 8; ++r) {
            float x  = acc[t][r] + bb;
            float sp = (x > 20.0f) ? x : log1pf(__expf(x));   // softplus
            dt[(size_t)(m0 + t * 16 + r + half * 8) * DI + col] = sp;
        }
}

// Zero-pad W_xproj (DI x 40) -> (DI x 48) so the x-proj GEMM needs no guards.
__global__ void k_pad_b(const float* __restrict__ src, float* __restrict__ dst)
{
    const int idx = blockIdx.x * blockDim.x + threadIdx.x;
    if (idx >= DI * DBL_LD) return;
    const int r = idx / DBL_LD, c = idx % DBL_LD;
    dst[idx] = (c < DBL_N) ? src[r * DBL_N + c] : 0.0f;
}

// Depthwise causal conv (k=4) + bias + SiLU. xi = xz[:, 0:DI], row stride 2*DI.
__global__ void k_conv_silu(const float* __restrict__ xz, const float* __restrict__ cw,
                            const float* __restrict__ cb, float* __restrict__ xc)
{
    const int idx = blockIdx.x * blockDim.x + threadIdx.x;
    if (idx >= L_SEQ * DI) return;
    const int d = idx % DI;
    const int l = idx / DI;
    float acc = cb[d];
    #pragma unroll
    for (int k = 0; k < DCONV; ++k) {
        const int ls = l - (DCONV - 1) + k;
        if (ls >= 0) acc += xz[ls * (2 * DI) + d] * cw[d * DCONV + k];
    }
    xc[idx] = acc / (1.0f + __expf(-acc));      // silu
}

// Selective scan: thread = one (d, s) recurrence; 16 blocks x 256 threads.
// h[l] = exp(dt*A)*h[l-1] + dt*xc*B[l,s]; y[l,d] = sum_s h*C[l,s] + Dp*xc,
// gated by silu(z). Reduction over s via 16-lane shfl_xor.
__global__ void k_scan(const float* __restrict__ dt, const float* __restrict__ xc,
                       const float* __restrict__ dbl, const float* __restrict__ xz,
                       const float* __restrict__ Alog, const float* __restrict__ Dskip,
                       float* __restrict__ yin)
{
    const int tid = threadIdx.x;
    const int s   = tid & 15;
    const int d   = blockIdx.x * 16 + (tid >> 4);
    const float A  = -__expf(Alog[d * DS + s]);
    const float Dp = Dskip[d];
    float h = 0.0f;
    for (int l = 0; l < L_SEQ; ++l) {
        const float dtv = dt[l * DI + d];
        const float xcv = xc[l * DI + d];
        const float Bv  = dbl[l * DBL_LD + DTR + s];
        const float Cv  = dbl[l * DBL_LD + DTR + DS + s];
        h = __expf(dtv * A) * h + (dtv * xcv) * Bv;
        float contrib = h * Cv;
        contrib += __shfl_xor(contrib, 8, 32);
        contrib += __shfl_xor(contrib, 4, 32);
        contrib += __shfl_xor(contrib, 2, 32);
        contrib += __shfl_xor(contrib, 1, 32);
        if (s == 0) {
            const float zv   = xz[l * (2 * DI) + DI + d];
            const float gate = zv / (1.0f + __expf(-zv));
            yin[l * DI + d] = (contrib + Dp * xcv) * gate;
        }
    }
}

// RMSNorm: one wave per token (128 values -> 4 per lane).
__global__ void k_rmsnorm(const float* __restrict__ y2, const float* __restrict__ w,
                          float* __restrict__ seq)
{
    const int wave = (blockIdx.x * blockDim.x + threadIdx.x) >> 5;
    const int lane = threadIdx.x & 31;
    if (wave >= L_SEQ) return;
    const float* row = y2 + (size_t)wave * DM;
    float v[4];
    float ss = 0.0f;
    #pragma unroll
    for (int i = 0; i < 4; ++i) { v[i] = row[lane * 4 + i]; ss += v[i] * v[i]; }
    ss += __shfl_xor(ss, 16, 32);
    ss += __shfl_xor(ss,  8, 32);
    ss += __shfl_xor(ss,  4, 32);
    ss += __shfl_xor(ss,  2, 32);
    ss += __shfl_xor(ss,  1, 32);
    const float scale = rsqrtf(ss / (float)DM + EPS_RMS);
    #pragma unroll
    for (int i = 0; i < 4; ++i)
        seq[(size_t)wave * DM + lane * 4 + i] = v[i] * scale * w[lane * 4 + i];
}

// (1,C,H,W) -> (L, C) and back.
__global__ void k_nchw_to_lc(const float* __restrict__ x, float* __restrict__ seq)
{
    const int idx = blockIdx.x * blockDim.x + threadIdx.x;
    if (idx >= L_SEQ * DM) return;
    const int c = idx % DM, p = idx / DM;
    seq[p * DM + c] = x[c * L_SEQ + p];
}
__global__ void k_lc_to_nchw(const float* __restrict__ seq, float* __restrict__ out)
{
    const int idx = blockIdx.x * blockDim.x + threadIdx.x;
    if (idx >= L_SEQ * DM) return;
    const int p = idx % L_SEQ, c = idx / L_SEQ;
    out[c * L_SEQ + p] = seq[p * DM + c];
}

extern "C" void kernel_launch(void* const* d_in, const int* in_sizes, int n_in,
                              void* d_out, int out_size, void* d_ws, size_t ws_size,
                              hipStream_t stream)
{
    (void)in_sizes; (void)n_in; (void)out_size; (void)ws_size;
    const float* x       = (const float*)d_in[0];
    const float* W_in    = (const float*)d_in[1];   // (8,128,512)
    const float* conv_w  = (const float*)d_in[2];   // (8,256,4)
    const float* conv_b  = (const float*)d_in[3];   // (8,256)
    const float* W_xproj = (const float*)d_in[4];   // (8,256,40)
    const float* W_dt    = (const float*)d_in[5];   // (8,8,256)
    const float* b_dt    = (const float*)d_in[6];   // (8,256)
    const float* A_log   = (const float*)d_in[7];   // (8,256,16)
    const float* D_skip  = (const float*)d_in[8];   // (8,256)
    const float* W_out   = (const float*)d_in[9];   // (8,256,128)
    const float* rms_w   = (const float*)d_in[10];  // (8,128)

    // Workspace carve-out (all counts multiples of 64 floats = 256B).
    float* ws  = (float*)d_ws;
    size_t off = 0;
    auto carve = [&](size_t n) { float* p = ws + off; off += (n + 63) & ~(size_t)63; return p; };
    float* seq  = carve((size_t)L_SEQ * DM);       // residual stream
    float* xz   = carve((size_t)L_SEQ * 2 * DI);   // in-proj output (xi | z)
    float* xc   = carve((size_t)L_SEQ * DI);       // conv+silu output
    float* dbl  = carve((size_t)L_SEQ * DBL_LD);   // [dt_r | B | C], padded to 48
    float* dt   = carve((size_t)L_SEQ * DI);       // softplus(dt)
    float* yin  = carve((size_t)L_SEQ * DI);       // gated scan output
    float* y2   = carve((size_t)L_SEQ * DM);       // pre-norm out-proj
    float* Wx48 = carve((size_t)DI * DBL_LD);      // zero-padded W_xproj

    constexpr int MT = 4;                          // M-tiles per wave
    constexpr int MG = MTILES / MT;                // 36 M-groups

    k_nchw_to_lc<<<(L_SEQ * DM + 255) / 256, 256, 0, stream>>>(x, seq);

    for (int l = 0; l < DEPTH; ++l) {
        const float* Wi  = W_in    + (size_t)l * DM * 2 * DI;
        const float* cw  = conv_w  + (size_t)l * DI * DCONV;
        const float* cb  = conv_b  + (size_t)l * DI;
        const float* Wx  = W_xproj + (size_t)l * DI * DBL_N;
        const float* Wdt = W_dt    + (size_t)l * DTR * DI;
        const float* bdt = b_dt    + (size_t)l * DI;
        const float* Al  = A_log   + (size_t)l * DI * DS;
        const float* Dp  = D_skip  + (size_t)l * DI;
        const float* Wo  = W_out   + (size_t)l * DI * DM;
        const float* rw  = rms_w   + (size_t)l * DM;

        // xz = seq @ W_in   (2304 x 512, K=128)
        {
            const int waves = MG * (2 * DI / 16);
            k_gemm<MT><<<(waves + 7) / 8, 256, 0, stream>>>(seq, Wi, xz,
                MG, 2 * DI / 16, DM, DM, 2 * DI, 2 * DI);
        }
        // xc = silu(causal_conv(xi) + cb)
        k_conv_silu<<<(L_SEQ * DI + 255) / 256, 256, 0, stream>>>(xz, cw, cb, xc);
        // pad W_xproj to N=48, then dbl = xc @ Wx48  (2304 x 48, K=256)
        k_pad_b<<<(DI * DBL_LD + 255) / 256, 256, 0, stream>>>(Wx, Wx48);
        {
            const int waves = MG * (DBL_LD / 16);
            k_gemm<MT><<<(waves + 7) / 8, 256, 0, stream>>>(xc, Wx48, dbl,
                MG, DBL_LD / 16, DI, DI, DBL_LD, DBL_LD);
        }
        // dt = softplus(dt_r @ W_dt + b_dt)   (2304 x 256, K=8)
        {
            const int waves = MG * (DI / 16);
            k_gemm_dt<MT><<<(waves + 7) / 8, 256, 0, stream>>>(dbl, Wdt, bdt, dt);
        }
        // selective scan + D-skip + silu(z) gating
        k_scan<<<DI / 16, 256, 0, stream>>>(dt, xc, dbl, xz, Al, Dp, yin);
        // y2 = yin @ W_out   (2304 x 128, K=256)
        {
            const int waves = MG * (DM / 16);
            k_gemm<MT><<<(waves + 7) / 8, 256, 0, stream>>>(yin, Wo, y2,
                MG, DM / 16, DI, DI, DM, DM);
        }
        // seq = rmsnorm(y2) * rms_w
        k_rmsnorm<<<(L_SEQ * 32 + 255) / 256, 256, 0, stream>>>(y2, rw, seq);
    }

    k_lc_to_nchw<<<(L_SEQ * DM + 255) / 256, 256, 0, stream>>>(seq, (float*)d_out);
}